// MoELayer_20358144983731
// MI455X (gfx1250) — compile-verified
//
#include <hip/hip_runtime.h>
#include <hip/hip_bf16.h>
#include <math.h>

// ---------------------------------------------------------------------------
// MoE per-type routed experts on gfx1250 (CDNA5), fp32 WMMA path.
//   x:[4,16384,256] f32, type_embeddings:[32,128], atom_types:[4,16384] i32,
//   gate_w:[128,8], expert_w:[8,256,256], expert_b:[8,256],
//   shared_w:[1,256,256], shared_b:[1,256]  ->  out:[4,16384,256] f32
// ---------------------------------------------------------------------------

#define DIN    256
#define DOUT   256
#define NTYPES 32
#define NEXP   8
#define TM     64          // tokens per workgroup tile
#define LDA    260         // padded LDS row stride (floats): 260 % 64 == 4 -> conflict-free

typedef __attribute__((ext_vector_type(2))) float v2f;
typedef __attribute__((ext_vector_type(8))) float v8f;

// builtin expects pointers to GCC-style int4 vectors (per hipcc diagnostic)
typedef int v4i_g __attribute__((vector_size(16)));
typedef __attribute__((address_space(1))) v4i_g* glds_gptr;
typedef __attribute__((address_space(3))) v4i_g* glds_lptr;

#if __has_builtin(__builtin_amdgcn_global_load_async_to_lds_b128) && \
    __has_builtin(__builtin_amdgcn_s_wait_asynccnt)
#define USE_ASYNC_LDS 1
#else
#define USE_ASYNC_LDS 0
#endif

__device__ __forceinline__ float fast_tanh(float v) {
#if __has_builtin(__builtin_amdgcn_tanhf)
    return __builtin_amdgcn_tanhf(v);     // V_TANH_F32 (gfx1250 trans op)
#else
    return tanhf(v);
#endif
}

// ---------------- kernel 1: gating (32 types x 8 experts) + zero counters ---
__global__ void moe_gate_kernel(const float* __restrict__ te,    // [32,128]
                                const float* __restrict__ gw,    // [128,8]
                                int* __restrict__ cnt,
                                int* __restrict__ e0t, int* __restrict__ e1t,
                                float* __restrict__ w0t, float* __restrict__ w1t) {
    int t = threadIdx.x;
    if (t >= NTYPES) return;
    cnt[t] = 0;
    float logit[NEXP];
    for (int e = 0; e < NEXP; ++e) {
        float s = 0.f;
        for (int k = 0; k < 128; ++k) s += te[t * 128 + k] * gw[k * NEXP + e];
        logit[e] = s;
    }
    // top-2 (ties -> lower index, matching jax.lax.top_k)
    int i0 = 0;
    for (int e = 1; e < NEXP; ++e) if (logit[e] > logit[i0]) i0 = e;
    int i1 = (i0 == 0) ? 1 : 0;
    for (int e = 0; e < NEXP; ++e) if (e != i0 && logit[e] > logit[i1]) i1 = e;
    float m  = logit[i0];
    float a  = __expf(logit[i0] - m);
    float b  = __expf(logit[i1] - m);
    float rs = 1.0f / (a + b);
    e0t[t] = i0; e1t[t] = i1;
    w0t[t] = a * rs; w1t[t] = b * rs;
}

// ---------------- kernel 2: histogram over token types ----------------------
__global__ void moe_hist_kernel(const int* __restrict__ types, int* __restrict__ cnt, int n) {
    int i = blockIdx.x * blockDim.x + threadIdx.x;
    if (i < n) atomicAdd(&cnt[types[i]], 1);
}

// ---------------- kernel 3: exclusive scan (32 buckets) ---------------------
__global__ void moe_scan_kernel(const int* __restrict__ cnt, int* __restrict__ off,
                                int* __restrict__ cur) {
    if (threadIdx.x == 0) {
        int s = 0;
        for (int t = 0; t < NTYPES; ++t) { off[t] = s; cur[t] = s; s += cnt[t]; }
    }
}

// ---------------- kernel 4: scatter token ids by type -----------------------
__global__ void moe_scatter_kernel(const int* __restrict__ types, int* __restrict__ cur,
                                   int* __restrict__ perm, int n) {
    int i = blockIdx.x * blockDim.x + threadIdx.x;
    if (i < n) {
        int pos = atomicAdd(&cur[types[i]], 1);
        perm[pos] = i;
    }
}

// ---------------- kernel 5: grouped expert GEMM with f32 WMMA ---------------
__global__ __launch_bounds__(256)
void moe_gemm_kernel(const float* __restrict__ x,         // [ntok, 256]
                     const float* __restrict__ expert_w,  // [8,256,256]
                     const float* __restrict__ expert_b,  // [8,256]
                     const float* __restrict__ shared_w,  // [256,256]
                     const float* __restrict__ shared_b,  // [256]
                     const int*   __restrict__ off,
                     const int*   __restrict__ cnt,
                     const int*   __restrict__ e0t, const int* __restrict__ e1t,
                     const float* __restrict__ w0t, const float* __restrict__ w1t,
                     const int*   __restrict__ perm,
                     float*       __restrict__ out) {
    __shared__ float sA[TM * LDA];
    __shared__ int   sTok[TM];

    const int g     = blockIdx.x;          // type group
    const int gcnt  = cnt[g];
    int       start = blockIdx.y * TM;
    if (start >= gcnt) return;
    const int rows  = min(TM, gcnt - start);
    start += off[g];

    const int tid = threadIdx.x;

    // padded rows replicate the last valid token of the tile (masked at store)
    if (tid < TM) sTok[tid] = perm[start + min(tid, rows - 1)];
    __syncthreads();

    // stage 64 gathered x-rows into LDS
    {
        const int r   = tid >> 2;                   // row 0..63
        const int q   = tid & 3;                    // 64-float quarter
        const int tok = sTok[r];
        const float4* src = (const float4*)(x + (size_t)tok * DIN + q * 64);
        float4*       dst = (float4*)(&sA[r * LDA + q * 64]);
#if USE_ASYNC_LDS
        #pragma unroll
        for (int i = 0; i < 16; ++i) {
            __builtin_amdgcn_global_load_async_to_lds_b128(
                (glds_gptr)(uintptr_t)(src + i),
                (glds_lptr)(uint32_t)(uintptr_t)(dst + i),
                0, 0);
        }
        __builtin_amdgcn_s_wait_asynccnt(0);
#else
        #pragma unroll
        for (int i = 0; i < 16; ++i) dst[i] = src[i];
#endif
    }
    __syncthreads();

    const int wave   = tid >> 5;
    const int lane   = tid & 31;
    const int laneLo = lane & 15;
    const int laneHi = lane >> 4;

    const int   e0 = e0t[g], e1 = e1t[g];
    const float w0 = w0t[g], w1 = w1t[g];

    for (int nt2 = 0; nt2 < 2; ++nt2) {
        const int n0 = (wave * 2 + nt2) * 16;

        float outacc[4][8];
        #pragma unroll
        for (int mt = 0; mt < 4; ++mt)
            #pragma unroll
            for (int r = 0; r < 8; ++r) outacc[mt][r] = 0.f;

        for (int ep = 0; ep < 3; ++ep) {
            const float* W; const float* B; float wgt;
            if (ep == 0)      { W = expert_w + (size_t)e0 * DIN * DOUT; B = expert_b + e0 * DOUT; wgt = w0;  }
            else if (ep == 1) { W = expert_w + (size_t)e1 * DIN * DOUT; B = expert_b + e1 * DOUT; wgt = w1;  }
            else              { W = shared_w;                           B = shared_b;             wgt = 1.f; }

            v8f acc[4];
            #pragma unroll
            for (int mt = 0; mt < 4; ++mt) acc[mt] = (v8f){};

            const float  bval = B[n0 + laneLo];
            const float* Wl   = W + n0 + laneLo;       // column n0+laneLo of W[k][n]

            // K = 256 in steps of 4 -> 64 chained v_wmma_f32_16x16x4_f32 per M-tile
            for (int k0 = 0; k0 < DIN; k0 += 4) {
                const int ks = k0 + 2 * laneHi;
                v2f bfrag;
                bfrag.x = Wl[(size_t)ks * DOUT];
                bfrag.y = Wl[(size_t)(ks + 1) * DOUT];
                #pragma unroll
                for (int mt = 0; mt < 4; ++mt) {
                    v2f afrag = *(const v2f*)(&sA[(mt * 16 + laneLo) * LDA + ks]);
                    acc[mt] = __builtin_amdgcn_wmma_f32_16x16x4_f32(
                        /*neg_a=*/false, afrag, /*neg_b=*/false, bfrag,
                        /*c_mod=*/(short)0, acc[mt],
                        /*reuse_a=*/false, /*reuse_b=*/false);
                }
            }

            #pragma unroll
            for (int mt = 0; mt < 4; ++mt)
                #pragma unroll
                for (int r = 0; r < 8; ++r)
                    outacc[mt][r] += wgt * fast_tanh(acc[mt][r] + bval);
        }

        // scatter stores: C/D layout -> N = n0+laneLo, M = r + 8*laneHi (+16*mt)
        #pragma unroll
        for (int mt = 0; mt < 4; ++mt) {
            #pragma unroll
            for (int r = 0; r < 8; ++r) {
                const int m = mt * 16 + r + 8 * laneHi;
                if (m < rows) {
                    const int tok = sTok[m];
                    out[(size_t)tok * DOUT + n0 + laneLo] = outacc[mt][r];
                }
            }
        }
    }
}

// ---------------------------------------------------------------------------
extern "C" void kernel_launch(void* const* d_in, const int* in_sizes, int n_in,
                              void* d_out, int out_size, void* d_ws, size_t ws_size,
                              hipStream_t stream) {
    const float* x        = (const float*)d_in[0];
    const float* te       = (const float*)d_in[1];
    const int*   types    = (const int*)  d_in[2];
    const float* gate_w   = (const float*)d_in[3];
    const float* expert_w = (const float*)d_in[4];
    const float* expert_b = (const float*)d_in[5];
    const float* shared_w = (const float*)d_in[6];
    const float* shared_b = (const float*)d_in[7];
    float*       out      = (float*)d_out;

    const int ntok = in_sizes[2];          // 65536

    // workspace layout
    int*   cnt  = (int*)d_ws;              // [32]
    int*   off  = cnt + NTYPES;            // [32]
    int*   cur  = off + NTYPES;            // [32]
    int*   e0t  = cur + NTYPES;            // [32]
    int*   e1t  = e0t + NTYPES;            // [32]
    float* w0t  = (float*)(e1t + NTYPES);  // [32]
    float* w1t  = w0t + NTYPES;            // [32]
    int*   perm = (int*)(w1t + NTYPES);    // [ntok]

    moe_gate_kernel<<<1, 32, 0, stream>>>(te, gate_w, cnt, e0t, e1t, w0t, w1t);

    const int tpb = 256;
    moe_hist_kernel<<<(ntok + tpb - 1) / tpb, tpb, 0, stream>>>(types, cnt, ntok);
    moe_scan_kernel<<<1, 32, 0, stream>>>(cnt, off, cur);
    moe_scatter_kernel<<<(ntok + tpb - 1) / tpb, tpb, 0, stream>>>(types, cur, perm, ntok);

    dim3 grid(NTYPES, (ntok + TM - 1) / TM);
    moe_gemm_kernel<<<grid, 256, 0, stream>>>(x, expert_w, expert_b, shared_w, shared_b,
                                              off, cnt, e0t, e1t, w0t, w1t, perm, out);
}